// Seq2Seq_73074573574149
// MI455X (gfx1250) — compile-verified
//
#include <hip/hip_runtime.h>
#include <math.h>

// Problem constants (from reference)
#define V_  5000
#define E_  256
#define H_  512
#define T_  128
#define B_  64
#define KC_ 128          // K-chunk staged in LDS
#define LP_ 132          // padded LDS row stride (132 % 64 == 4 -> conflict-free)

typedef __attribute__((ext_vector_type(2))) float v2f;
typedef __attribute__((ext_vector_type(8))) float v8f;

// CDNA5 fp32 WMMA: D(16x16,f32) = A(16x4,f32) x B(4x16,f32) + C
#define WMMA_F32(acc, a, b) \
    acc = __builtin_amdgcn_wmma_f32_16x16x4_f32(false, (a), false, (b), (short)0, (acc), false, false)

__device__ __forceinline__ float sigf(float x) { return 1.0f / (1.0f + __expf(-x)); }

// ---------------------------------------------------------------------------
// Zero-fill helper (state init + output row 0)
// ---------------------------------------------------------------------------
__global__ void zero_kernel(float* __restrict__ p, long n) {
    long i = (long)blockIdx.x * blockDim.x + threadIdx.x;
    if (i < n) p[i] = 0.0f;
}

// ---------------------------------------------------------------------------
// One LSTM layer step:  g = x @ Wih.T + h_in @ Whh.T + b ; cell update.
// grid.x = H/16 = 32 (16-col slice of H), block = 512 (16 waves = 4 gates x 4
// batch row blocks). A fragments (x / h rows) are staged through a padded LDS
// K-chunk (cooperative global->LDS, then ds_load_b64 per WMMA step), removing
// the 4x gate-wave redundancy on activation reads. B fragments (weight rows,
// unique per workgroup, L2-resident) load straight from global.
// ---------------------------------------------------------------------------
__global__ __launch_bounds__(512)
void lstm_layer_kernel(const float* __restrict__ Wih, int Kin,
                       const float* __restrict__ Whh,
                       const float* __restrict__ bias,
                       const float* __restrict__ emb,
                       const int*   __restrict__ toks,
                       const float* __restrict__ x_glb,
                       const float* __restrict__ h_in,
                       float*       __restrict__ c,      // B x H, in/out (in place)
                       float*       __restrict__ h_out)  // B x H
{
    __shared__ float chunk[B_][LP_];     // staged A rows, one K-chunk (~34 KB)
    __shared__ float g_lds[4][B_][16];   // gate exchange (16 KB)

    const int wave = threadIdx.x >> 5;
    const int lane = threadIdx.x & 31;
    const int gate = wave >> 2;          // 0..3 : i,f,g,o
    const int rb   = wave & 3;           // batch row block 0..3
    const int jb   = blockIdx.x;         // 16-col slice of H
    const int ln   = lane & 15;
    const int hi   = lane >> 4;          // 0: K0/K1 half, 1: K2/K3 half
    const int row  = rb * 16 + ln;       // batch row this lane feeds for A
    const int wrow = gate * H_ + jb * 16 + ln;  // weight row this lane feeds for B

    // cooperative loader mapping: 64 rows x 128 cols = 16 floats per thread
    const int lr  = threadIdx.x >> 3;    // row 0..63
    const int seg = threadIdx.x & 7;     // 16-float segment 0..7

    v8f acc = {};

    for (int phase = 0; phase < 2; ++phase) {
        const int    Ksrc  = phase ? H_  : Kin;
        const float* Wbase = phase ? Whh : Wih;
        const float* wr    = Wbase + (long)wrow * Ksrc;

        // this thread's source row for the cooperative stage
        const float* srow;
        if (phase)          srow = h_in + (long)lr * H_;
        else if (emb)       srow = emb + (long)toks[lr] * Kin;
        else                srow = x_glb + (long)lr * Kin;

        for (int k0 = 0; k0 < Ksrc; k0 += KC_) {
            __syncthreads();             // previous chunk fully consumed
            {
                const float4* sp = (const float4*)(srow + k0 + seg * 16);
                float4 t0 = sp[0], t1 = sp[1], t2 = sp[2], t3 = sp[3];
                float4* dp = (float4*)&chunk[lr][seg * 16];
                dp[0] = t0; dp[1] = t1; dp[2] = t2; dp[3] = t3;
            }
            __syncthreads();             // chunk visible to all waves

            if (k0 + KC_ < Ksrc)         // pull next weight chunk toward L1/L2
                __builtin_prefetch(wr + k0 + KC_, 0, 1);

            #pragma unroll 8
            for (int kk = 0; kk < KC_; kk += 4) {
                v2f a = *(const v2f*)&chunk[row][kk + 2 * hi];
                v2f b = *(const v2f*)(wr + k0 + kk + 2 * hi);
                WMMA_F32(acc, a, b);
            }
        }
    }

    // ---- exchange gates through LDS (C layout: N = ln, M = v + 8*hi) ----
    #pragma unroll
    for (int v = 0; v < 8; ++v)
        g_lds[gate][rb * 16 + hi * 8 + v][ln] = acc[v];
    __syncthreads();

    // ---- fused cell update: 64 rows x 16 cols = 1024 elems / 512 threads ----
    for (int e = threadIdx.x; e < B_ * 16; e += 512) {
        const int b_  = e >> 4;
        const int n   = e & 15;
        const int col = jb * 16 + n;
        const float iv = g_lds[0][b_][n] + bias[0 * H_ + col];
        const float fv = g_lds[1][b_][n] + bias[1 * H_ + col];
        const float gv = g_lds[2][b_][n] + bias[2 * H_ + col];
        const float ov = g_lds[3][b_][n] + bias[3 * H_ + col];
        const long  ix = (long)b_ * H_ + col;
        const float c_new = sigf(fv) * c[ix] + sigf(iv) * tanhf(gv);
        c[ix]     = c_new;
        h_out[ix] = sigf(ov) * tanhf(c_new);
    }
}

// ---------------------------------------------------------------------------
// Decoder output projection, fused with vocab_sim multiplier:
//   out[t] = (h1 @ fc_W.T + fc_b) * mult(src[t-1], src[t])
// grid.x = ceil(V/64) = 79, block = 512 (16 waves = 4 row blocks x 4 col
// subtiles). h1 is staged through the LDS K-chunk once per WG (4x fewer
// cross-WG h1 re-reads than a 16-col-per-WG layout). Epilogue runs directly
// on the WMMA accumulator registers.
// ---------------------------------------------------------------------------
__global__ __launch_bounds__(512)
void fc_kernel(const float* __restrict__ fc_W,
               const float* __restrict__ fc_b,
               const float* __restrict__ h1,        // B x H
               const float* __restrict__ vocab_sim, // V x V
               const int*   __restrict__ src,       // T x B
               int t,
               float*       __restrict__ out)       // T x B x V
{
    __shared__ float chunk[B_][LP_];

    const int wave = threadIdx.x >> 5;
    const int lane = threadIdx.x & 31;
    const int rb   = wave & 3;           // batch row block
    const int ns   = wave >> 2;          // 16-col subtile within the 64-col slice
    const int ln   = lane & 15;
    const int hi   = lane >> 4;
    const int col0 = blockIdx.x * 64 + ns * 16;
    const int row  = rb * 16 + ln;
    const int wrow = min(col0 + ln, V_ - 1);   // clamp for the partial last tile

    const int lr  = threadIdx.x >> 3;
    const int seg = threadIdx.x & 7;

    v8f acc = {};
    const float* wr = fc_W + (long)wrow * H_;

    for (int k0 = 0; k0 < H_; k0 += KC_) {
        __syncthreads();
        {
            const float4* sp = (const float4*)(h1 + (long)lr * H_ + k0 + seg * 16);
            float4 t0 = sp[0], t1 = sp[1], t2 = sp[2], t3 = sp[3];
            float4* dp = (float4*)&chunk[lr][seg * 16];
            dp[0] = t0; dp[1] = t1; dp[2] = t2; dp[3] = t3;
        }
        __syncthreads();

        if (k0 + KC_ < H_)
            __builtin_prefetch(wr + k0 + KC_, 0, 1);

        #pragma unroll 8
        for (int kk = 0; kk < KC_; kk += 4) {
            v2f a = *(const v2f*)&chunk[row][kk + 2 * hi];
            v2f b = *(const v2f*)(wr + k0 + kk + 2 * hi);
            WMMA_F32(acc, a, b);
        }
    }

    const int col = col0 + ln;                 // this lane's output column
    if (col < V_) {
        const float bcol = fc_b[col];
        #pragma unroll
        for (int v = 0; v < 8; ++v) {
            const int   b_ = rb * 16 + hi * 8 + v;
            const int   s0 = src[(long)(t - 1) * B_ + b_];
            const int   s1 = src[(long)t * B_ + b_];
            float sim = fminf(vocab_sim[(long)s0 * V_ + col],
                              vocab_sim[(long)s1 * V_ + col]);
            float md  = fminf(sim, 20.0f);
            if (col == 3 || col == 4) md = 0.0f;
            float mult = md * md - 20.0f * md + 1.0f;
            if (col == V_ - 1) mult = 1.0f;
            out[((long)t * B_ + b_) * (long)V_ + col] = (acc[v] + bcol) * mult;
        }
    }
}

// ---------------------------------------------------------------------------
// Host orchestration: per-timestep kernel chain on `stream`.
// Workspace layout (floats): h0a h0b c0 h1a h1b c1   (6 * B*H)
// ---------------------------------------------------------------------------
extern "C" void kernel_launch(void* const* d_in, const int* in_sizes, int n_in,
                              void* d_out, int out_size, void* d_ws, size_t ws_size,
                              hipStream_t stream) {
    const int*   src       = (const int*)  d_in[0];
    const int*   trg       = (const int*)  d_in[2];
    const float* vocab_sim = (const float*)d_in[4];
    const float* emb_enc   = (const float*)d_in[5];
    const float* emb_dec   = (const float*)d_in[6];
    const float* enc_Wih0  = (const float*)d_in[7];
    const float* enc_Whh0  = (const float*)d_in[8];
    const float* enc_b0    = (const float*)d_in[9];
    const float* enc_Wih1  = (const float*)d_in[10];
    const float* enc_Whh1  = (const float*)d_in[11];
    const float* enc_b1    = (const float*)d_in[12];
    const float* dec_Wih0  = (const float*)d_in[13];
    const float* dec_Whh0  = (const float*)d_in[14];
    const float* dec_b0    = (const float*)d_in[15];
    const float* dec_Wih1  = (const float*)d_in[16];
    const float* dec_Whh1  = (const float*)d_in[17];
    const float* dec_b1    = (const float*)d_in[18];
    const float* fc_W      = (const float*)d_in[19];
    const float* fc_b      = (const float*)d_in[20];
    float* out = (float*)d_out;

    const long BH = (long)B_ * H_;
    float* ws  = (float*)d_ws;
    float* h0a = ws + 0 * BH;
    float* h0b = ws + 1 * BH;
    float* c0  = ws + 2 * BH;
    float* h1a = ws + 3 * BH;
    float* h1b = ws + 4 * BH;
    float* c1  = ws + 5 * BH;

    // Zero state (6*BH) and output timestep 0 (B*V).
    {
        long n = 6 * BH;
        zero_kernel<<<(unsigned)((n + 255) / 256), 256, 0, stream>>>(ws, n);
        long m = (long)B_ * V_;
        zero_kernel<<<(unsigned)((m + 255) / 256), 256, 0, stream>>>(out, m);
    }

    float *h0c = h0a, *h0n = h0b, *h1c = h1a, *h1n = h1b;
    const unsigned gridL = H_ / 16;           // 32
    const unsigned gridF = (V_ + 63) / 64;    // 79

    // -------- Encoder: T steps --------
    for (int t = 0; t < T_; ++t) {
        lstm_layer_kernel<<<gridL, 512, 0, stream>>>(
            enc_Wih0, E_, enc_Whh0, enc_b0,
            emb_enc, src + (long)t * B_, nullptr,
            h0c, c0, h0n);
        lstm_layer_kernel<<<gridL, 512, 0, stream>>>(
            enc_Wih1, H_, enc_Whh1, enc_b1,
            nullptr, nullptr, h0n,
            h1c, c1, h1n);
        float* tmp;
        tmp = h0c; h0c = h0n; h0n = tmp;
        tmp = h1c; h1c = h1n; h1n = tmp;
    }

    // -------- Decoder: steps t = 1 .. T-1 --------
    for (int t = 1; t < T_; ++t) {
        lstm_layer_kernel<<<gridL, 512, 0, stream>>>(
            dec_Wih0, E_, dec_Whh0, dec_b0,
            emb_dec, trg + (long)(t - 1) * B_, nullptr,
            h0c, c0, h0n);
        lstm_layer_kernel<<<gridL, 512, 0, stream>>>(
            dec_Wih1, H_, dec_Whh1, dec_b1,
            nullptr, nullptr, h0n,
            h1c, c1, h1n);
        fc_kernel<<<gridF, 512, 0, stream>>>(
            fc_W, fc_b, h1n, vocab_sim, src, t, out);
        float* tmp;
        tmp = h0c; h0c = h0n; h0n = tmp;
        tmp = h1c; h1c = h1n; h1n = tmp;
    }
}